// GCNConv_77257871720698
// MI455X (gfx1250) — compile-verified
//
#include <hip/hip_runtime.h>

#define N_NODES 100000
#define N_EDGES 3200000
#define D 128          // D_IN == D_OUT == 128

typedef __attribute__((ext_vector_type(2))) float v2f;
typedef __attribute__((ext_vector_type(8))) float v8f;
typedef __attribute__((ext_vector_type(4))) unsigned int u32x4;
typedef __attribute__((ext_vector_type(8))) int i32x8;
typedef __attribute__((ext_vector_type(4))) int i32x4;

static_assert(N_EDGES % 256 == 0, "scatter kernel assumes 256 edges per block");

// ---------------------------------------------------------------------------
// Kernel 1: zero the workspace (agg[N*D] ++ deg[N]) with 16B stores.
// total floats = 100000*128 + 100000 = 12,900,000 (divisible by 4).
// ---------------------------------------------------------------------------
__global__ void gcn_zero_ws(float4* __restrict__ ws, int nvec4) {
    int i = blockIdx.x * blockDim.x + threadIdx.x;
    if (i < nvec4) {
        ws[i] = make_float4(0.f, 0.f, 0.f, 0.f);
    }
}

// ---------------------------------------------------------------------------
// Kernel 2: edge scatter. Each wave32 owns 32 edges: every lane loads one
// (src,dst) pair coalesced and bumps deg[dst] once (non-divergent, one atomic
// per edge). Then an unrolled loop broadcasts each edge via v_readlane and
// the whole wave gathers x[src] with one b128 load per lane and scatters
// 4 no-return global_atomic_add_f32 into agg[dst]. Working set (x 51MB +
// agg 51MB) lives in the 192MB L2, so these atomics resolve at L2 rate.
// ---------------------------------------------------------------------------
__global__ void __launch_bounds__(256)
gcn_scatter_edges(const float* __restrict__ x,
                  const int* __restrict__ esrc,
                  const int* __restrict__ edst,
                  float* __restrict__ agg,
                  float* __restrict__ deg) {
    int wave = blockIdx.x * 8 + (threadIdx.x >> 5);   // 8 waves per block
    int lane = threadIdx.x & 31;
    int eBase = wave * 32;                            // 32 edges per wave

    int mySrc = esrc[eBase + lane];                   // coalesced index loads
    int myDst = edst[eBase + lane];

    atomicAdd(deg + myDst, 1.0f);                     // one deg bump per edge

#pragma unroll
    for (int j = 0; j < 32; ++j) {
        int s = __shfl(mySrc, j, 32);                 // v_readlane broadcast
        int d = __shfl(myDst, j, 32);
        float4 v = ((const float4*)(x + (size_t)s * D))[lane]; // 512B/wave
        float* arow = agg + (size_t)d * D + lane * 4;
        atomicAdd(arow + 0, v.x);
        atomicAdd(arow + 1, v.y);
        atomicAdd(arow + 2, v.z);
        atomicAdd(arow + 3, v.w);
    }
}

// ---------------------------------------------------------------------------
// Kernel 3: out = (agg / max(deg,1)) @ W + bias using V_WMMA_F32_16X16X4_F32.
//
// W[128][128] is DMA'd into LDS by the Tensor Data Mover (wave 0 issues one
// tensor_load_to_lds). The D# enables LDS padding (4 DWORDs every 128 DWORDs)
// so rows land at stride 132 floats -> the two B-fragment reads per step hit
// distinct banks (stride 128 would be a 16-way conflict on 64 banks).
//
// D# layout per CDNA5 ISA:
//   group0: count=1 | lds_addr | global_addr | type=2
//   group1: data_size=2(4B), pad_enable=1, pad_interval=6(128dw),
//           pad_amount=3(4dw), tensor_dim0=128, tensor_dim1=128,
//           tile_dim0=128, tile_dim1=128, tensor_dim0_stride=128
//   groups 2/3: zero (2-D tensor)
//
// WMMA fragment layouts per CDNA5 ISA §7.12.2:
//   A 16x4 f32 : lanes 0-15 -> M=lane, K={k0,k0+1}; lanes16-31 -> K={k0+2,k0+3}
//   B 4x16 f32 : lanes 0-15 -> K={k0,k0+1} at N=lane; lanes16-31 -> K={k0+2,k0+3}
//   C/D 16x16  : VGPR r: lanes 0-15 -> M=r, N=lane; lanes16-31 -> M=r+8
// ---------------------------------------------------------------------------
__global__ void __launch_bounds__(256)
gcn_wmma_gemm(const float* __restrict__ agg,
              const float* __restrict__ deg,
              const float* __restrict__ W,      // [K=128][N=128] row-major
              const float* __restrict__ bias,   // [128]
              float* __restrict__ out) {        // [N_NODES][128]
    constexpr int LDW = D + 4;                  // LDS row stride produced by TDM pad
    __shared__ float sW[D * LDW];

    int tid = threadIdx.x;

    if (tid < 32) {                             // wave 0 issues the TDM descriptor
        unsigned lds_base = (unsigned)(uintptr_t)(&sW[0]);   // LDS byte offset
        unsigned long long ga = (unsigned long long)(uintptr_t)W;

        u32x4 g0;
        g0[0] = 1u;                                          // count=1 (valid user D#)
        g0[1] = lds_base;                                    // lds_addr [63:32]
        g0[2] = (unsigned)(ga & 0xffffffffu);                // global_addr [95:64]
        g0[3] = (unsigned)((ga >> 32) & 0x01ffffffu)         // global_addr [120:96]
              | (2u << 30);                                  // type=2 ("image")

        i32x8 g1;
        g1[0] = (2 << 16)    // data_size = 4 bytes
              | (1 << 20)    // pad_enable
              | (6 << 22)    // pad_interval: 128 DWORDs
              | (3 << 25);   // pad_amount:   4 DWORDs  -> stride 132 floats
        g1[1] = (D << 16);   // tensor_dim0[15:0] in bits [63:48]
        g1[2] = (D << 16);   // tensor_dim0 hi = 0; tensor_dim1[15:0] in [95:80]
        g1[3] = (D << 16);   // tensor_dim1 hi = 0; tile_dim0 = 128 in [127:112]
        g1[4] = D;           // tile_dim1 = 128 in [143:128]; tile_dim2 = 0
        g1[5] = D;           // tensor_dim0_stride[31:0] = 128
        g1[6] = 0;           // stride hi / tensor_dim1_stride lo
        g1[7] = 0;

        i32x4 z4 = {0, 0, 0, 0};                 // groups 2/3 unused (2-D tensor)
        i32x8 z8 = {0, 0, 0, 0, 0, 0, 0, 0};     // extra group (6-arg builtin form)
        __builtin_amdgcn_tensor_load_to_lds(g0, g1, z4, z4, z8, /*cpol=*/0);
        __builtin_amdgcn_s_wait_tensorcnt(0);                // s_wait_tensorcnt 0
    }
    __syncthreads();                                         // LDS now holds padded W

    int wave = tid >> 5;
    int lane = tid & 31;
    int half = lane >> 4;          // 0: lanes 0-15, 1: lanes 16-31
    int lm   = lane & 15;
    int rowBase = blockIdx.x * 128 + wave * 16;

    int myRow = rowBase + lm;
    int rowClamped = (myRow < N_NODES) ? myRow : 0;    // keep EXEC all-1 for WMMA
    float scale = 1.0f / fmaxf(deg[rowClamped], 1.0f); // fuse mean into A
    const float* arow = agg + (size_t)rowClamped * D;
    int koff = half * 2;

    v8f c[8] = {};                 // 8 N-tiles of 16 -> N = 128

    for (int k0 = 0; k0 < D; k0 += 4) {
        v2f a = *(const v2f*)(arow + k0 + koff);       // 8B load, K pair
        a *= scale;
#pragma unroll
        for (int nt = 0; nt < 8; ++nt) {
            int n = nt * 16 + lm;
            v2f b;
            b.x = sW[(k0 + koff + 0) * LDW + n];
            b.y = sW[(k0 + koff + 1) * LDW + n];
            c[nt] = __builtin_amdgcn_wmma_f32_16x16x4_f32(
                /*neg_a=*/false, a, /*neg_b=*/false, b,
                /*c_mod=*/(short)0, c[nt],
                /*reuse_a=*/false, /*reuse_b=*/false);
        }
    }

    // Epilogue: + bias, store per C/D layout. The row-bound check is
    // wave-uniform except in the one tail block -> single scalar branch.
    if (rowBase + 16 <= N_NODES) {
#pragma unroll
        for (int nt = 0; nt < 8; ++nt) {
            int n = nt * 16 + lm;
            float bn = bias[n];
#pragma unroll
            for (int r = 0; r < 8; ++r) {
                out[(size_t)(rowBase + r + half * 8) * D + n] = c[nt][r] + bn;
            }
        }
    } else {
#pragma unroll
        for (int nt = 0; nt < 8; ++nt) {
            int n = nt * 16 + lm;
            float bn = bias[n];
#pragma unroll
            for (int r = 0; r < 8; ++r) {
                int m = rowBase + r + half * 8;
                if (m < N_NODES) out[(size_t)m * D + n] = c[nt][r] + bn;
            }
        }
    }
}

// ---------------------------------------------------------------------------
// Launch. Inputs in setup_inputs() order:
//   d_in[0] x[N,128] f32, d_in[1] weight[128,128] f32, d_in[2] bias[128] f32,
//   d_in[3] edge_src[E] i32, d_in[4] edge_dst[E] i32.
// d_ws layout: agg (N*D f32) | deg (N f32)  -> ~51.6 MB.
// ---------------------------------------------------------------------------
extern "C" void kernel_launch(void* const* d_in, const int* in_sizes, int n_in,
                              void* d_out, int out_size, void* d_ws, size_t ws_size,
                              hipStream_t stream) {
    const float* x    = (const float*)d_in[0];
    const float* W    = (const float*)d_in[1];
    const float* bias = (const float*)d_in[2];
    const int*  esrc  = (const int*)d_in[3];
    const int*  edst  = (const int*)d_in[4];
    float* out = (float*)d_out;

    float* agg = (float*)d_ws;
    float* deg = agg + (size_t)N_NODES * D;

    // 1) zero agg + deg
    int nvec4 = (N_NODES * D + N_NODES) / 4;
    gcn_zero_ws<<<(nvec4 + 255) / 256, 256, 0, stream>>>((float4*)d_ws, nvec4);

    // 2) edge scatter: 32 edges per wave, 8 waves per block
    int sblocks = N_EDGES / 256;
    gcn_scatter_edges<<<sblocks, 256, 0, stream>>>(x, esrc, edst, agg, deg);

    // 3) WMMA GEMM + mean + bias: 128 rows per block
    int gblocks = (N_NODES + 127) / 128;
    gcn_wmma_gemm<<<gblocks, 256, 0, stream>>>(agg, deg, W, bias, out);
}